// GLiClassUniEncoder_979252544165
// MI455X (gfx1250) — compile-verified
//
#include <hip/hip_runtime.h>
#include <hip/hip_bf16.h>

#define Bn 8
#define Sn 2048
#define Hn 1024
#define MAXC 16

typedef __attribute__((ext_vector_type(2))) float v2f;
typedef __attribute__((ext_vector_type(8))) float v8f;

// ---------------- meta: per-row scan --------------------------------------
// meta layout (ints): [0..127] cls_pos[8][16], [128..255] cls_valid[8][16],
// [256..263] text_start, [264..271] tlen, [272..279] den(f32), [280..287] bscale(f32)
__global__ void gliclass_meta_kernel(const int* __restrict__ ids,
                                     const int* __restrict__ attn,
                                     int* __restrict__ meta,
                                     float* __restrict__ out_mask) {
    const int b = blockIdx.x;
    const int tid = threadIdx.x;
    __shared__ unsigned bits[Sn / 32];
    __shared__ int s_ts, s_eos, s_cnt;
    if (tid < Sn / 32) bits[tid] = 0u;
    if (tid == 0) { s_ts = Sn; s_eos = -1; s_cnt = 0; }
    __syncthreads();
    for (int p = tid; p < Sn; p += blockDim.x) {
        const int id = ids[b * Sn + p];
        if (id == 1) atomicOr(&bits[p >> 5], 1u << (p & 31));
        if (id == 2) atomicMin(&s_ts, p);
        if (attn[b * Sn + p] != 0) atomicMax(&s_eos, p);
    }
    __syncthreads();
    const int ts   = (s_ts == Sn) ? 0 : s_ts;          // argmax of all-false = 0
    const int eos  = (s_eos < 0) ? (Sn - 1) : s_eos;   // all-zero attn -> S-1
    const int tlen = eos - ts;
    int cnt = 0;
    for (int t = tid; t < tlen; t += blockDim.x)
        cnt += (attn[b * Sn + ts + t] != 0) ? 1 : 0;
    atomicAdd(&s_cnt, cnt);
    __syncthreads();
    if (tid == 0) {
        int* cls_pos = meta;
        int* cls_val = meta + 128;
        int nc_total = 0, nc = 0;
        for (int w = 0; w < Sn / 32; ++w) {
            unsigned x = bits[w];
            nc_total += __popc(x);
            while (x && nc < MAXC) {
                const int bit = __ffs(x) - 1;
                cls_pos[b * MAXC + nc] = w * 32 + bit;
                ++nc;
                x &= x - 1u;
            }
        }
        for (int c = 0; c < MAXC; ++c) {
            const int v = (c < nc_total) ? 1 : 0;
            cls_val[b * MAXC + c] = v;
            out_mask[b * MAXC + c] = (float)v;
            if (c >= nc) cls_pos[b * MAXC + c] = 0;
        }
        meta[256 + b] = ts;
        meta[264 + b] = tlen;
        const float den = (float)s_cnt;
        ((float*)(meta + 272))[b] = den;
        ((float*)(meta + 280))[b] = den / fmaxf(den, 1e-8f);
    }
}

// ---------------- pool: masked mean of text span --------------------------
__global__ void gliclass_pool_kernel(const float* __restrict__ emb,
                                     const int* __restrict__ attn,
                                     const int* __restrict__ meta,
                                     float* __restrict__ meanpad) {
    const int brow = blockIdx.x;                       // 0..15 (rows >=8 are zero pad)
    const int h = blockIdx.y * blockDim.x + threadIdx.x;
    if (brow >= Bn) { meanpad[brow * Hn + h] = 0.f; return; }
    const int ts = meta[256 + brow];
    const int tlen = meta[264 + brow];
    const float den = ((const float*)(meta + 272))[brow];
    const float* ep = emb + ((size_t)brow * Sn + ts) * Hn + h;
    const int* ap = attn + brow * Sn + ts;
    float a0 = 0.f, a1 = 0.f, a2 = 0.f, a3 = 0.f;
    int t = 0;
    for (; t + 3 < tlen; t += 4) {
        a0 += ep[(size_t)(t + 0) * Hn] * (float)ap[t + 0];
        a1 += ep[(size_t)(t + 1) * Hn] * (float)ap[t + 1];
        a2 += ep[(size_t)(t + 2) * Hn] * (float)ap[t + 2];
        a3 += ep[(size_t)(t + 3) * Hn] * (float)ap[t + 3];
    }
    for (; t < tlen; ++t) a0 += ep[(size_t)t * Hn] * (float)ap[t];
    const float acc = (a0 + a1) + (a2 + a3);
    meanpad[brow * Hn + h] = acc / fmaxf(den, 1e-8f);
}

// ---------------- gather class-token embeddings ---------------------------
__global__ void gliclass_gather_kernel(const float* __restrict__ emb,
                                       const int* __restrict__ meta,
                                       float* __restrict__ cls_out) {
    const int idx = blockIdx.x;                        // b*16 + c
    const int b = idx >> 4, c = idx & 15;
    const int pos = meta[b * MAXC + c];
    const int val = meta[128 + b * MAXC + c];
    const float* src = emb + ((size_t)b * Sn + pos) * Hn;
    float* dst = cls_out + (size_t)idx * Hn;
    for (int h = threadIdx.x; h < Hn; h += blockDim.x)
        dst[h] = val ? src[h] : 0.f;
}

// ---------------- WMMA f32 16x16xK tile (one wave, EXEC all ones) ---------
__device__ __forceinline__ v8f wmma_f32_tile(const float* __restrict__ A, int lda,
                                             const float* __restrict__ Bm, int ldb,
                                             int m0, int n0, int K) {
    const int lane = threadIdx.x & 31;
    const int half = lane >> 4;                        // K offset selector
    const int l16 = lane & 15;
    v8f c = {};
    const float* arow = A + (size_t)(m0 + l16) * lda;  // lane = M row
    for (int k0 = 0; k0 < K; k0 += 4) {
        const int ka = k0 + 2 * half;
        v2f a = *(const v2f*)(arow + ka);              // A[m][ka], A[m][ka+1]
        v2f bb;
        bb.x = Bm[(size_t)(ka + 0) * ldb + n0 + l16];  // B[ka][n],  lane = N col
        bb.y = Bm[(size_t)(ka + 1) * ldb + n0 + l16];  // B[ka+1][n]
        c = __builtin_amdgcn_wmma_f32_16x16x4_f32(false, a, false, bb,
                                                  (short)0, c, false, false);
    }
    return c;
}

// ---------------- pooled = meanpad @ Wt + bt*bscale -----------------------
__global__ void gliclass_pooled_gemm_kernel(const float* __restrict__ meanpad,
                                            const float* __restrict__ Wt,
                                            const float* __restrict__ bt,
                                            const int* __restrict__ meta,
                                            float* __restrict__ pooled) {
    const int n0 = blockIdx.x * 16;
    v8f c = wmma_f32_tile(meanpad, Hn, Wt, Hn, 0, n0, Hn);
    const int lane = threadIdx.x & 31;
    const int half = lane >> 4;
    const int l16 = lane & 15;
    const float* bsc = (const float*)(meta + 280);
    const float bias = bt[n0 + l16];
#pragma unroll
    for (int r = 0; r < 8; ++r) {
        const int m = r + 8 * half;                    // D: VGPR r -> M=r (+8 upper half)
        if (m < Bn)
            pooled[m * Hn + n0 + l16] = c[r] + bias * bsc[m];
    }
}

// ---------------- cls_proj = cls_emb @ Wc + bc ----------------------------
__global__ void gliclass_cls_gemm_kernel(const float* __restrict__ cls_emb,
                                         const float* __restrict__ Wc,
                                         const float* __restrict__ bc,
                                         float* __restrict__ cls_proj) {
    const int n0 = blockIdx.x * 16;
    const int m0 = blockIdx.y * 16;
    v8f c = wmma_f32_tile(cls_emb, Hn, Wc, Hn, m0, n0, Hn);
    const int lane = threadIdx.x & 31;
    const int half = lane >> 4;
    const int l16 = lane & 15;
    const float bias = bc[n0 + l16];
#pragma unroll
    for (int r = 0; r < 8; ++r) {
        const int m = m0 + r + 8 * half;
        cls_proj[(size_t)m * Hn + n0 + l16] = c[r] + bias;
    }
}

// ---------------- logits[b,c] = pooled[b] . cls_proj[b,c] -----------------
__global__ void gliclass_logits_kernel(const float* __restrict__ pooled,
                                       const float* __restrict__ cls_proj,
                                       const int* __restrict__ meta,
                                       const float* __restrict__ logit_scale,
                                       float* __restrict__ logits) {
    const int wid = (blockIdx.x * blockDim.x + threadIdx.x) >> 5;
    const int lane = threadIdx.x & 31;
    if (wid >= Bn * MAXC) return;
    const int b = wid >> 4;
    const float* p = pooled + b * Hn;
    const float* q = cls_proj + (size_t)wid * Hn;
    float acc = 0.f;
    for (int h = lane; h < Hn; h += 32) acc += p[h] * q[h];
    for (int m = 16; m >= 1; m >>= 1) acc += __shfl_xor(acc, m, 32);
    if (lane == 0) {
        const int valid = meta[128 + wid];
        logits[wid] = valid ? acc * logit_scale[0] : 0.f;
    }
}

extern "C" void kernel_launch(void* const* d_in, const int* in_sizes, int n_in,
                              void* d_out, int out_size, void* d_ws, size_t ws_size,
                              hipStream_t stream) {
    (void)in_sizes; (void)n_in; (void)out_size; (void)ws_size;
    const float* emb   = (const float*)d_in[0];
    const int*   ids   = (const int*)d_in[1];
    const int*   attn  = (const int*)d_in[2];
    const float* Wc    = (const float*)d_in[3];
    const float* bc    = (const float*)d_in[4];
    const float* Wt    = (const float*)d_in[5];
    const float* bt    = (const float*)d_in[6];
    const float* lsc   = (const float*)d_in[7];

    float* out        = (float*)d_out;
    float* out_logits = out;                              // [8*16]
    float* out_cls    = out + Bn * MAXC;                  // [8*16*1024]
    float* out_mask   = out + Bn * MAXC + Bn * MAXC * Hn; // [8*16]

    float* ws       = (float*)d_ws;
    float* meanpad  = ws;                                 // [16*1024] (zero-padded)
    float* pooled   = ws + 16 * Hn;                       // [8*1024]
    float* cls_proj = ws + 24 * Hn;                       // [128*1024]
    int*   meta     = (int*)(ws + 24 * Hn + Bn * MAXC * Hn); // 288 ints

    gliclass_meta_kernel<<<Bn, 256, 0, stream>>>(ids, attn, meta, out_mask);
    dim3 gpool(16, Hn / 256);
    gliclass_pool_kernel<<<gpool, 256, 0, stream>>>(emb, attn, meta, meanpad);
    gliclass_gather_kernel<<<Bn * MAXC, 256, 0, stream>>>(emb, meta, out_cls);
    gliclass_pooled_gemm_kernel<<<Hn / 16, 32, 0, stream>>>(meanpad, Wt, bt, meta, pooled);
    dim3 gcls(Hn / 16, (Bn * MAXC) / 16);
    gliclass_cls_gemm_kernel<<<gcls, 32, 0, stream>>>(out_cls, Wc, bc, cls_proj);
    gliclass_logits_kernel<<<(Bn * MAXC * 32 + 127) / 128, 128, 0, stream>>>(
        pooled, cls_proj, meta, lsc, out_logits);
}